// ContinuousHMM_15719580303713
// MI455X (gfx1250) — compile-verified
//
#include <hip/hip_runtime.h>
#include <math.h>

typedef __attribute__((ext_vector_type(2))) float v2f;
typedef __attribute__((ext_vector_type(8))) float v8f;

#define NSEQ 32768
#define TLEN 128
#define DDIM 32
#define PDIM 16
#define NBIN 15

// ---------------------------------------------------------------------------
// Phase 1: gd[n*T+t] = L_dyn[n,t,:] . w_gamma_dyn ; bd = L_dyn[n,t,:] . w_beta_dyn
// Implemented as (rows,32) x (32,2) GEMV with V_WMMA_F32_16X16X4_F32 chains.
// Exact fp32 multiply-accumulate; memory bound (streams 536 MB of L_dyn).
// ---------------------------------------------------------------------------
__global__ void dyn_dots_wmma(const float* __restrict__ L,
                              const float* __restrict__ wg,
                              const float* __restrict__ wb,
                              float* __restrict__ gd,
                              float* __restrict__ bd,
                              int ntiles)
{
    const int lane   = threadIdx.x & 31;
    const int wave   = (blockIdx.x * blockDim.x + threadIdx.x) >> 5;
    const int nwaves = (gridDim.x * blockDim.x) >> 5;
    const int r = lane & 15;   // A: row in tile; B/D: column index N
    const int h = lane >> 4;   // half-wave select (K split)

    // B fragments for the eight K-chunks: 4x16 tile, col0 = w_gamma, col1 = w_beta.
    // Layout: lanes 0-15 hold K={4c+0,4c+1}, lanes 16-31 hold K={4c+2,4c+3}.
    v2f bfrag[8];
#pragma unroll
    for (int c = 0; c < 8; ++c) {
        int k = 4 * c + 2 * h;
        float g0 = wg[k], g1 = wg[k + 1];
        float b0 = wb[k], b1 = wb[k + 1];
        bfrag[c].x = (r == 0) ? g0 : ((r == 1) ? b0 : 0.0f);
        bfrag[c].y = (r == 0) ? g1 : ((r == 1) ? b1 : 0.0f);
    }

    for (int tile = wave; tile < ntiles; tile += nwaves) {
        // A fragment source: lane covers row (tile*16 + r), K base 2*h, step 4.
        const float* Arow = L + (size_t)(tile * 16 + r) * DDIM + 2 * h;
        v8f acc = {};
#pragma unroll
        for (int c = 0; c < 8; ++c) {
            v2f a = *(const v2f*)(Arow + 4 * c);  // 8-byte aligned (even offsets)
            acc = __builtin_amdgcn_wmma_f32_16x16x4_f32(
                false, a, false, bfrag[c], (short)0, acc, false, false);
        }
        // D layout: VGPR v, lanes 0-15 -> D[M=v][N=lane]; lanes 16-31 -> D[M=v+8][N=lane-16]
        int mbase = tile * 16 + 8 * h;
        if (r == 0) {
#pragma unroll
            for (int v = 0; v < 8; ++v) gd[mbase + v] = acc[v];
        } else if (r == 1) {
#pragma unroll
            for (int v = 0; v < 8; ++v) bd[mbase + v] = acc[v];
        }
    }
}

// ---------------------------------------------------------------------------
// Phase 2: per-sequence static dots (tiny: 2 MB)
// ---------------------------------------------------------------------------
__global__ void static_dots(const float* __restrict__ C,
                            const float* __restrict__ wgs,
                            const float* __restrict__ wbs,
                            float* __restrict__ gs,
                            float* __restrict__ bs,
                            int n_total)
{
    int n = blockIdx.x * blockDim.x + threadIdx.x;
    if (n >= n_total) return;
    const float4* c4 = (const float4*)(C + (size_t)n * PDIM);
    float g = 0.0f, b = 0.0f;
#pragma unroll
    for (int i = 0; i < 4; ++i) {
        float4 cv = c4[i];
        g += cv.x * wgs[4*i+0] + cv.y * wgs[4*i+1] + cv.z * wgs[4*i+2] + cv.w * wgs[4*i+3];
        b += cv.x * wbs[4*i+0] + cv.y * wbs[4*i+1] + cv.z * wbs[4*i+2] + cv.w * wbs[4*i+3];
    }
    gs[n] = g;
    bs[n] = b;
}

// ---------------------------------------------------------------------------
// Phase 3: sequential Kalman/Laplace scan, one thread per sequence.
// Each thread streams 2.5 KB contiguously (S,Y,mask,gd,bd via float4 loads).
// ---------------------------------------------------------------------------
__device__ __forceinline__ void scan_step(
    float s, float y, float m, float gdv, float bdv, float ts,
    float psi, float gA, float sigma_sq, float b0, float bZ, float bT,
    float gsn, float bsn, const float* bins,
    float& Zm, float& Zvv, float& ll, float& zf_out, float& zv_out)
{
    // bucketize: idx = count(inner_edges < s), inner edge i = -2.5 + (i+1)/3
    float u = (s + 2.5f) * 3.0f;
    int idx = (int)ceilf(u) - 1;
    idx = min(max(idx, 0), NBIN - 1);
    float be = bins[idx];

    float Zpm = psi * Zm + gA * s + gdv + gsn;
    float Zpv = psi * psi * Zvv + sigma_sq;
    float logit = b0 + bZ * Zpm + be + bdv + bsn + bT * ts;
    logit = fminf(fmaxf(logit, -20.0f), 20.0f);
    float prob = 1.0f / (1.0f + expf(-logit));
    float grad = (y - prob) * bZ * m;
    float hess = bZ * bZ * prob * (1.0f - prob) * m + 1e-6f;
    float Zpost_v = 1.0f / (1.0f / (Zpv + 1e-8f) + hess);
    Zm  = Zpm + Zpost_v * grad;
    Zvv = Zpost_v;
    ll += (y * logf(prob + 1e-10f) + (1.0f - y) * logf(1.0f - prob + 1e-10f)) * m;
    zf_out = Zm;
    zv_out = Zvv;
}

__global__ void kalman_scan(const float* __restrict__ S,
                            const float* __restrict__ Y,
                            const float* __restrict__ mask,
                            const float* __restrict__ gd,
                            const float* __restrict__ bd,
                            const float* __restrict__ gs,
                            const float* __restrict__ bs,
                            const float* __restrict__ psi_p,
                            const float* __restrict__ gA_p,
                            const float* __restrict__ lsZ_p,
                            const float* __restrict__ b0_p,
                            const float* __restrict__ bZ_p,
                            const float* __restrict__ bins_g,
                            const float* __restrict__ bT_p,
                            float* __restrict__ Zf,
                            float* __restrict__ Zv,
                            float* __restrict__ ll_partials,
                            int n_total)
{
    __shared__ float red[256];
    __shared__ float bins[NBIN];
    if (threadIdx.x < NBIN) bins[threadIdx.x] = bins_g[threadIdx.x];
    __syncthreads();

    const int n = blockIdx.x * blockDim.x + threadIdx.x;
    float ll = 0.0f;

    if (n < n_total) {
        const float psi = psi_p[0];
        const float gA  = gA_p[0];
        const float sigma_sq = expf(2.0f * lsZ_p[0]);   // exp(log_sigma)^2
        const float b0  = b0_p[0];
        const float bZ  = bZ_p[0];
        const float bT  = bT_p[0];
        const float gsn = gs[n];
        const float bsn = bs[n];

        const float4* S4  = (const float4*)(S    + (size_t)n * TLEN);
        const float4* Y4  = (const float4*)(Y    + (size_t)n * TLEN);
        const float4* M4  = (const float4*)(mask + (size_t)n * TLEN);
        const float4* G4  = (const float4*)(gd   + (size_t)n * TLEN);
        const float4* B4  = (const float4*)(bd   + (size_t)n * TLEN);
        float4* ZF4 = (float4*)(Zf + (size_t)n * TLEN);
        float4* ZV4 = (float4*)(Zv + (size_t)n * TLEN);

        float Zm = 0.0f, Zvv = 1.0f;

        for (int q = 0; q < TLEN / 4; ++q) {
            float4 s4 = S4[q], y4 = Y4[q], m4 = M4[q], g4 = G4[q], d4 = B4[q];
            float4 zf4, zv4;
            int t0 = 4 * q;
            scan_step(s4.x, y4.x, m4.x, g4.x, d4.x, (float)(t0 + 0) / 30.0f,
                      psi, gA, sigma_sq, b0, bZ, bT, gsn, bsn, bins,
                      Zm, Zvv, ll, zf4.x, zv4.x);
            scan_step(s4.y, y4.y, m4.y, g4.y, d4.y, (float)(t0 + 1) / 30.0f,
                      psi, gA, sigma_sq, b0, bZ, bT, gsn, bsn, bins,
                      Zm, Zvv, ll, zf4.y, zv4.y);
            scan_step(s4.z, y4.z, m4.z, g4.z, d4.z, (float)(t0 + 2) / 30.0f,
                      psi, gA, sigma_sq, b0, bZ, bT, gsn, bsn, bins,
                      Zm, Zvv, ll, zf4.z, zv4.z);
            scan_step(s4.w, y4.w, m4.w, g4.w, d4.w, (float)(t0 + 3) / 30.0f,
                      psi, gA, sigma_sq, b0, bZ, bT, gsn, bsn, bins,
                      Zm, Zvv, ll, zf4.w, zv4.w);
            ZF4[q] = zf4;
            ZV4[q] = zv4;
        }
    }

    // Deterministic fixed-order block reduction of ll
    red[threadIdx.x] = ll;
    __syncthreads();
    for (int s = 128; s > 0; s >>= 1) {
        if (threadIdx.x < s) red[threadIdx.x] += red[threadIdx.x + s];
        __syncthreads();
    }
    if (threadIdx.x == 0) ll_partials[blockIdx.x] = red[0];
}

// Final deterministic reduction of per-block partials (single block).
__global__ void ll_reduce(const float* __restrict__ partials,
                          float* __restrict__ out, int nparts)
{
    __shared__ float red[256];
    int tid = threadIdx.x;
    float v = 0.0f;
    for (int i = tid; i < nparts; i += 256) v += partials[i];
    red[tid] = v;
    __syncthreads();
    for (int s = 128; s > 0; s >>= 1) {
        if (tid < s) red[tid] += red[tid + s];
        __syncthreads();
    }
    if (tid == 0) out[0] = red[0];
}

// ---------------------------------------------------------------------------
extern "C" void kernel_launch(void* const* d_in, const int* in_sizes, int n_in,
                              void* d_out, int out_size, void* d_ws, size_t ws_size,
                              hipStream_t stream) {
    const float* S     = (const float*)d_in[0];
    const float* L_dyn = (const float*)d_in[1];
    const float* C_st  = (const float*)d_in[2];
    const float* Y     = (const float*)d_in[3];
    const float* mask  = (const float*)d_in[4];
    const float* psi   = (const float*)d_in[5];
    const float* gA    = (const float*)d_in[6];
    const float* w_gd  = (const float*)d_in[7];
    const float* w_gs  = (const float*)d_in[8];
    const float* lsZ   = (const float*)d_in[9];
    const float* b0    = (const float*)d_in[10];
    const float* bZ    = (const float*)d_in[11];
    const float* bins  = (const float*)d_in[12];
    const float* w_bd  = (const float*)d_in[13];
    const float* w_bs  = (const float*)d_in[14];
    const float* bT    = (const float*)d_in[15];

    const int n_total = in_sizes[0] / TLEN;      // N (S has N*T elements)
    const int rows    = n_total * TLEN;          // N*T

    float* out = (float*)d_out;
    float* Zf = out;
    float* Zv = out + (size_t)rows;
    float* ll = out + (size_t)2 * rows;

    // workspace layout
    float* ws  = (float*)d_ws;
    float* gd  = ws;                                  // N*T
    float* bd  = gd + (size_t)rows;                   // N*T
    float* gs  = bd + (size_t)rows;                   // N
    float* bs  = gs + (size_t)n_total;                // N
    float* llp = bs + (size_t)n_total;                // block partials

    // Phase 1: WMMA GEMV over L_dyn (memory-bound streaming pass)
    const int ntiles = rows / 16;
    dyn_dots_wmma<<<1024, 256, 0, stream>>>(L_dyn, w_gd, w_bd, gd, bd, ntiles);

    // Phase 2: static dots
    static_dots<<<(n_total + 255) / 256, 256, 0, stream>>>(C_st, w_gs, w_bs, gs, bs, n_total);

    // Phase 3: sequential scan, one thread per sequence
    const int scan_blocks = (n_total + 255) / 256;
    kalman_scan<<<scan_blocks, 256, 0, stream>>>(
        S, Y, mask, gd, bd, gs, bs,
        psi, gA, lsZ, b0, bZ, bins, bT,
        Zf, Zv, llp, n_total);

    // Phase 4: deterministic final ll reduction
    ll_reduce<<<1, 256, 0, stream>>>(llp, ll, scan_blocks);
}